// ObjectLoss_43379169690270
// MI455X (gfx1250) — compile-verified
//
#include <hip/hip_runtime.h>

// ObjectLoss: segmented weighted mean -> scalar.
//   num[p] = sum_i mse_i*xi_i [pid_i==p & rec_i], den[p] = sum_i xi_i [...]
//   out    = mean_{p=1..P-1}(num[p]/den[p])
// Memory-bound: 88 MB streamed -> ~3.8 us floor @ 23.3 TB/s. 400 KB accumulator
// lives in L2; f32 atomics are non-returning (STOREcnt path).

#define P_SEG 50000

typedef __attribute__((ext_vector_type(2))) float v2f;
typedef __attribute__((ext_vector_type(8))) float v8f;

__device__ __forceinline__ float lane_gather(float v, int srcLane) {
  // ds_bpermute: index in bytes
  return __int_as_float(__builtin_amdgcn_ds_bpermute(srcLane << 2, __float_as_int(v)));
}

__global__ void __launch_bounds__(256) zero_f32(float* __restrict__ p, int n) {
  int i = blockIdx.x * blockDim.x + threadIdx.x;
  if (i < n) p[i] = 0.0f;
}

// 128 threads = 4 wave32 per block; each wave handles 32 hits.
__global__ void __launch_bounds__(128) hit_scatter(
    const float* __restrict__ beta, const float* __restrict__ pred,
    const int* __restrict__ pid, const float* __restrict__ tp,
    const int* __restrict__ rec, float* __restrict__ num,
    float* __restrict__ den, int n) {
  __shared__ float lds[4 * 512];  // per-wave: 2 x (32 lanes x 8 dwords) D-tiles
  const int lane = threadIdx.x & 31;
  const int wave = threadIdx.x >> 5;
  const int base = blockIdx.x * 128 + (wave << 5);
  const int i = base + lane;

  float mse = 0.0f;
  bool active;
  if (base + 32 <= n) {   // full wave: EXEC all-1s -> WMMA legal
    active = true;
    const float4 p = ((const float4*)pred)[i];
    const float4 t = ((const float4*)tp)[i];
    const float d0 = p.x - t.x, d1 = p.y - t.y, d2 = p.z - t.z, d3 = p.w - t.w;
    const int sl = lane & 15;

    // A-matrix 16x4 f32 layout: lanes 0-15 hold (K0,K1) of row M=lane,
    // lanes 16-31 hold (K2,K3) of row M=lane-16. Build two tiles:
    //   tile1 = hits base..base+15, tile2 = hits base+16..base+31.
    float lo0 = lane_gather(d0, sl),      lo1 = lane_gather(d1, sl);
    float hi0 = lane_gather(d2, sl),      hi1 = lane_gather(d3, sl);
    v2f A1; A1.x = (lane < 16) ? lo0 : hi0; A1.y = (lane < 16) ? lo1 : hi1;
    lo0 = lane_gather(d0, 16 + sl);       lo1 = lane_gather(d1, 16 + sl);
    hi0 = lane_gather(d2, 16 + sl);       hi1 = lane_gather(d3, 16 + sl);
    v2f A2; A2.x = (lane < 16) ? lo0 : hi0; A2.y = (lane < 16) ? lo1 : hi1;

    // Same registers as A and B: B layout mirrors A's striping, so this
    // computes the Gram matrix diff * diff^T; diagonal = per-hit mse.
    v8f c1 = {}; v8f c2 = {};
    c1 = __builtin_amdgcn_wmma_f32_16x16x4_f32(false, A1, false, A1,
                                               (short)0, c1, false, false);
    c2 = __builtin_amdgcn_wmma_f32_16x16x4_f32(false, A2, false, A2,
                                               (short)0, c2, false, false);

    // Diagonal extraction via LDS (DS ops are in-order within a wave).
    float* wl1 = &lds[wave * 512];
    float* wl2 = wl1 + 256;
#pragma unroll
    for (int r = 0; r < 8; ++r) {
      wl1[lane * 8 + r] = c1[r];
      wl2[lane * 8 + r] = c2[r];
    }
    __builtin_amdgcn_wave_barrier();  // keep compiler from reordering DS ops
    // C/D 16x16 f32 layout: VGPR r, lanes 0-15: (M=r, N=lane); lanes 16-31:
    // (M=r+8, N=lane-16). Diagonal (h,h): h<8 -> (vgpr h, lane h),
    // h>=8 -> (vgpr h-8, lane h+16).
    const int h  = lane & 15;
    const int rr = (h < 8) ? h : (h - 8);
    const int LL = (h < 8) ? h : (h + 16);
    const float diag1 = wl1[LL * 8 + rr];  // mse of hit base+h   (lanes 0-15)
    const float diag2 = wl2[LL * 8 + rr];  // mse of hit base+16+h (lanes 16-31)
    mse = (lane < 16) ? diag1 : diag2;
  } else {                 // ragged tail: plain VALU fallback
    active = (i < n);
    if (active) {
      const float4 p = ((const float4*)pred)[i];
      const float4 t = ((const float4*)tp)[i];
      const float d0 = p.x - t.x, d1 = p.y - t.y, d2 = p.z - t.z, d3 = p.w - t.w;
      mse = d0 * d0 + d1 * d1 + d2 * d2 + d3 * d3;
    }
  }

  if (active) {
    const float b  = beta[i];
    const float a  = 0.5f * __logf((1.0f + b) / (1.0f - b));  // arctanh
    const float xi = a * a;
    const int   id = pid[i];
    if (rec[i] > 0 && id > 0) {       // bin 0 is dropped by the reference
      atomicAdd(&num[id], mse * xi);  // non-returning global_atomic_add_f32
      atomicAdd(&den[id], xi);
    }
  }
}

__global__ void __launch_bounds__(256) reduce_mean_ratio(
    const float* __restrict__ num, const float* __restrict__ den,
    float* __restrict__ out) {
  __shared__ float s[256];
  float acc = 0.0f;
  for (int p = 1 + (int)threadIdx.x; p < P_SEG; p += 256)
    acc += num[p] / den[p];
  s[threadIdx.x] = acc;
  __syncthreads();
  for (int off = 128; off > 0; off >>= 1) {
    if ((int)threadIdx.x < off) s[threadIdx.x] += s[threadIdx.x + off];
    __syncthreads();
  }
  if (threadIdx.x == 0) out[0] = s[0] / (float)(P_SEG - 1);
}

extern "C" void kernel_launch(void* const* d_in, const int* in_sizes, int n_in,
                              void* d_out, int out_size, void* d_ws, size_t ws_size,
                              hipStream_t stream) {
  (void)n_in; (void)out_size; (void)ws_size;
  const float* beta = (const float*)d_in[0];
  const float* pred = (const float*)d_in[1];
  const int*   pid  = (const int*)d_in[2];
  const float* tp   = (const float*)d_in[3];
  const int*   rec  = (const int*)d_in[4];
  float* num = (float*)d_ws;          // P_SEG floats
  float* den = num + P_SEG;           // P_SEG floats (400 KB total, L2-resident)
  float* out = (float*)d_out;
  const int n = in_sizes[0];          // N hits

  zero_f32<<<(2 * P_SEG + 255) / 256, 256, 0, stream>>>(num, 2 * P_SEG);
  hit_scatter<<<(n + 127) / 128, 128, 0, stream>>>(beta, pred, pid, tp, rec,
                                                   num, den, n);
  reduce_mean_ratio<<<1, 256, 0, stream>>>(num, den, out);
}